// Decode_Map_30794915512909
// MI455X (gfx1250) — compile-verified
//
#include <hip/hip_runtime.h>
#include <stdint.h>

// Problem constants (from reference): heatmap (32, 80, 128, 128) fp32, K=100.
#define BS   32
#define CC   80
#define HH   128
#define WW   128
#define HW   (HH * WW)       // 16384
#define KTOP 100
#define NEG  (-3.402823466e38f)

// ---------------------------------------------------------------------------
// CDNA5 async global->LDS copy (GLOBAL_LOAD_ASYNC_TO_LDS_B128, ASYNCcnt).
// GVS addressing: mem = SGPR64 + VGPR32 offset; VDST VGPR holds LDS byte addr.
// ---------------------------------------------------------------------------
__device__ __forceinline__ void async_ld_b128(uint32_t lds_off, const void* base,
                                              uint32_t goff) {
  asm volatile("global_load_async_to_lds_b128 %0, %1, %2"
               :
               : "v"(lds_off), "v"(goff), "s"(base)
               : "memory");
}
__device__ __forceinline__ void wait_async0() {
  asm volatile("s_wait_asynccnt 0" ::: "memory");
}

// Wave32 argmax reduction: (value desc, index asc) + carried owner-thread id.
__device__ __forceinline__ void argmax_reduce(float& v, int& gi, int& wt) {
#pragma unroll
  for (int m = 16; m; m >>= 1) {
    float ov = __shfl_xor(v, m, 32);
    int   og = __shfl_xor(gi, m, 32);
    int   ot = __shfl_xor(wt, m, 32);
    if (ov > v || (ov == v && og < gi)) { v = ov; gi = og; wt = ot; }
  }
}

// ---------------------------------------------------------------------------
// Kernel 1: one block per (b,c) slice. Async-stage slice to LDS, fused 3x3 NMS,
// then iterative top-100 extraction with cooperative winner-wave rescans.
// Writes per-class scores to d_out region 0 and spatial indices to d_ws.
// ---------------------------------------------------------------------------
__global__ __launch_bounds__(256) void nms_topk_class(const float* __restrict__ hm,
                                                      float* __restrict__ outScores,
                                                      int* __restrict__ wsIdx) {
  __shared__ float S[HW];        // 64 KB slice (CDNA5: up to 320 KB LDS / WG)
  __shared__ float lmax[256];
  __shared__ int   lidx[256];
  __shared__ float s_wv;
  __shared__ int   s_wi, s_wt;

  const int t  = threadIdx.x;
  const int bc = blockIdx.x;     // b*80 + c

  // ---- async stage: 256 threads x 16 x b128 = 64 KB ----
  const uint32_t ldsBase = (uint32_t)(uintptr_t)(&S[0]) + (uint32_t)t * 256u;
  const uint32_t gBase   = (uint32_t)bc * (uint32_t)(HW * 4) + (uint32_t)t * 256u;
#pragma unroll
  for (int j = 0; j < 16; ++j)
    async_ld_b128(ldsBase + (uint32_t)j * 16u, hm, gBase + (uint32_t)j * 16u);
  wait_async0();
  __syncthreads();

  // ---- fused 3x3 max-pool NMS: thread t owns row y = t>>1, cols c0..c0+63 ----
  const int y  = t >> 1;
  const int c0 = (t & 1) * 64;
  const int rb = y * WW;
  float r[64];
  {
    auto vmax3 = [&](int c) -> float {
      float m = S[rb + c];
      if (y > 0)      m = fmaxf(m, S[rb - WW + c]);
      if (y < HH - 1) m = fmaxf(m, S[rb + WW + c]);
      return m;
    };
    float a = (c0 > 0) ? vmax3(c0 - 1) : NEG;   // rolling 3-column window
    float b = vmax3(c0);
#pragma unroll
    for (int j = 0; j < 64; ++j) {
      const int c   = c0 + j;
      const float cn = (c < WW - 1) ? vmax3(c + 1) : NEG;
      const float m  = fmaxf(a, fmaxf(b, cn));
      const float v  = S[rb + c];
      r[j] = (m <= v) ? v : 0.0f;               // keep peaks, zero the rest
      a = b; b = cn;
    }
  }
  __syncthreads();                              // all reads done before overwrite

  // ---- write back suppressed values (segment = S[t*64 .. t*64+63]) ----
  {
    float bv = NEG; int bi = t * 64;
#pragma unroll
    for (int j = 0; j < 64; ++j) {
      S[t * 64 + j] = r[j];
      if (r[j] > bv) { bv = r[j]; bi = t * 64 + j; }   // strict > keeps lowest idx
    }
    lmax[t] = bv; lidx[t] = bi;
  }
  __syncthreads();

  // ---- iterative top-100 extraction ----
  for (int it = 0; it < KTOP; ++it) {
    if (t < 32) {                               // wave0: block argmax over 256
      float v = NEG; int gi = 0x7fffffff; int wt = 0;
#pragma unroll
      for (int s = 0; s < 8; ++s) {
        const int u = t + s * 32;
        const float ov = lmax[u]; const int og = lidx[u];
        if (ov > v || (ov == v && og < gi)) { v = ov; gi = og; wt = u; }
      }
      argmax_reduce(v, gi, wt);
      if (t == 0) { s_wv = v; s_wi = gi; s_wt = wt; }
    }
    __syncthreads();
    const float bv = s_wv; const int bi = s_wi; const int bt = s_wt;
    if (t == 0) {
      outScores[bc * KTOP + it] = bv;           // (BS,C,K) scores -> out region 0
      wsIdx[bc * KTOP + it]     = bi;           // spatial index for kernel 2
    }
    if (t == bt) S[bi] = NEG;                   // remove extracted element
    __syncthreads();
    if ((t >> 5) == (bt >> 5)) {                // winner's wave rescans its 64-seg
      const int lane = t & 31;
      const int base = bt * 64;
      float v = S[base + lane];       int gi = base + lane;
      const float v1 = S[base + 32 + lane]; const int g1 = base + 32 + lane;
      if (v1 > v) { v = v1; gi = g1; }          // strict >: lower idx wins ties
      int wt = bt;
      argmax_reduce(v, gi, wt);
      if (lane == 0) { lmax[bt] = v; lidx[bt] = gi; }
    }
    __syncthreads();
  }
}

// ---------------------------------------------------------------------------
// Kernel 2: one block per batch. Top-100 over the C*K = 8000 per-class scores,
// then emit inds / classes / ys / xs (as floats) to out regions 1..4.
// ---------------------------------------------------------------------------
__global__ __launch_bounds__(256) void topk_batch(const float* __restrict__ scores,
                                                  const int* __restrict__ wsIdx,
                                                  float* __restrict__ out) {
  __shared__ float Cd[8192];                    // 8000 candidates, padded
  __shared__ float lmax[256];
  __shared__ int   lidx[256];
  __shared__ int   s_wi, s_wt;

  const int t = threadIdx.x;
  const int b = blockIdx.x;
  const float* src = scores + b * (CC * KTOP);

  for (int i = t; i < 8192; i += 256)
    Cd[i] = (i < CC * KTOP) ? src[i] : NEG;
  __syncthreads();

  {
    float bv = NEG; int bi = t * 32;
#pragma unroll
    for (int j = 0; j < 32; ++j) {
      const float v = Cd[t * 32 + j];
      if (v > bv) { bv = v; bi = t * 32 + j; }
    }
    lmax[t] = bv; lidx[t] = bi;
  }
  __syncthreads();

  const int o_inds = BS * CC * KTOP;            // 256000
  const int o_cls  = o_inds + BS * KTOP;        // 259200
  const int o_ys   = o_cls + BS * KTOP;         // 262400
  const int o_xs   = o_ys + BS * KTOP;          // 265600

  for (int it = 0; it < KTOP; ++it) {
    if (t < 32) {
      float v = NEG; int gi = 0x7fffffff; int wt = 0;
#pragma unroll
      for (int s = 0; s < 8; ++s) {
        const int u = t + s * 32;
        const float ov = lmax[u]; const int og = lidx[u];
        if (ov > v || (ov == v && og < gi)) { v = ov; gi = og; wt = u; }
      }
      argmax_reduce(v, gi, wt);
      if (t == 0) { s_wi = gi; s_wt = wt; }
    }
    __syncthreads();
    const int bi = s_wi; const int bt = s_wt;
    if (t == 0) {
      const int ci = bi;                        // index into (C*K)
      const int sp = wsIdx[b * (CC * KTOP) + ci];
      out[o_inds + b * KTOP + it] = (float)sp;
      out[o_cls  + b * KTOP + it] = (float)(ci / KTOP);
      out[o_ys   + b * KTOP + it] = (float)(sp >> 7);        // / 128
      out[o_xs   + b * KTOP + it] = (float)(sp & (WW - 1));  // % 128
    }
    if (t == bt) Cd[bi] = NEG;
    __syncthreads();
    if ((t >> 5) == (bt >> 5)) {                // winner wave rescans 32-seg
      const int lane = t & 31;
      float v = Cd[bt * 32 + lane]; int gi = bt * 32 + lane;
      int wt = bt;
      argmax_reduce(v, gi, wt);
      if (lane == 0) { lmax[bt] = v; lidx[bt] = gi; }
    }
    __syncthreads();
  }
}

// ---------------------------------------------------------------------------
extern "C" void kernel_launch(void* const* d_in, const int* in_sizes, int n_in,
                              void* d_out, int out_size, void* d_ws, size_t ws_size,
                              hipStream_t stream) {
  (void)in_sizes; (void)n_in; (void)out_size; (void)ws_size;
  const float* hm  = (const float*)d_in[0];
  float*       out = (float*)d_out;
  int*         wsi = (int*)d_ws;                // 256000 ints (1 MB) of scratch

  nms_topk_class<<<BS * CC, 256, 0, stream>>>(hm, out, wsi);
  topk_batch<<<BS, 256, 0, stream>>>(out, wsi, out);
}